// IGCLGNNModel_36421322670671
// MI455X (gfx1250) — compile-verified
//
#include <hip/hip_runtime.h>
#include <hip/hip_bf16.h>

// ---------------------------------------------------------------------------
// GIN (4-layer) inference for MI455X / gfx1250, fp32 end-to-end.
// GEMMs use native fp32 WMMA: v_wmma_f32_16x16x4_f32 (wave32).
// v2: transposed-W LDS layout -> B fragments load as single ds_load_b64
//     (conflict-free, no VGPR shuffle movs); 32 rows/wave doubles B reuse.
// ---------------------------------------------------------------------------

#define N_NODES   50000
#define N_EDGES   800000
#define N_GRAPHS  512
#define D         128
#define L_LAYERS  4
#define NUM_CLS   10
#define BN_EPS    1e-5f

// transposed weight-chunk LDS: sWt[col * WT_STRIDE + k], k in [0,128)
// stride 132 == 4 (mod 64 banks) => b64 fragment reads hit all 64 banks once
#define WT_STRIDE 132

typedef float v2f __attribute__((ext_vector_type(2)));
typedef float v8f __attribute__((ext_vector_type(8)));

__device__ __forceinline__ void atomAddF(float* p, float v) {
    unsafeAtomicAdd(p, v);   // lowers to global_atomic_add_f32
}

// ---------------------------------------------------------------------------
// Utility kernels
// ---------------------------------------------------------------------------
__global__ void zero_kernel(float* __restrict__ p, int n) {
    int i = blockIdx.x * blockDim.x + threadIdx.x;
    if (i < n) p[i] = 0.f;
}

// t[i] = h[i]  (seeds the "+h" of GIN before edge scatter)
__global__ void copy4_kernel(const float* __restrict__ src, float* __restrict__ dst, int n4) {
    int i = blockIdx.x * blockDim.x + threadIdx.x;
    if (i < n4) ((float4*)dst)[i] = ((const float4*)src)[i];
}

// Edge scatter: t[dst] += h[src].  One wave-lane handles 4 contiguous dims.
__global__ void edge_scatter_kernel(const int* __restrict__ e_src,
                                    const int* __restrict__ e_dst,
                                    const float* __restrict__ h,
                                    float* __restrict__ t) {
    long long tid = (long long)blockIdx.x * blockDim.x + threadIdx.x;
    long long edge = tid >> 5;           // 32 lanes per edge
    if (edge >= N_EDGES) return;
    int lane = (int)(tid & 31);
    int s = e_src[edge];
    int d = e_dst[edge];
    const float4 hv = *(const float4*)(h + (size_t)s * D + lane * 4);
    float* out = t + (size_t)d * D + lane * 4;
    atomAddF(out + 0, hv.x);
    atomAddF(out + 1, hv.y);
    atomAddF(out + 2, hv.z);
    atomAddF(out + 3, hv.w);
}

// Global sum-pool into feat[:, layer*D : (layer+1)*D]
__global__ void pool_kernel(const float* __restrict__ h,
                            const int* __restrict__ batch,
                            float* __restrict__ feat, int layer) {
    long long tid = (long long)blockIdx.x * blockDim.x + threadIdx.x;
    long long node = tid >> 5;
    if (node >= N_NODES) return;
    int lane = (int)(tid & 31);
    int g = batch[node];
    const float4 hv = *(const float4*)(h + (size_t)node * D + lane * 4);
    float* out = feat + (size_t)g * (L_LAYERS * D) + layer * D + lane * 4;
    atomAddF(out + 0, hv.x);
    atomAddF(out + 1, hv.y);
    atomAddF(out + 2, hv.z);
    atomAddF(out + 3, hv.w);
}

// ---------------------------------------------------------------------------
// WMMA GEMM:  out[M,128] = act( BN( A[M,K] @ W[K,128] + bias ) )
//   - 4 waves / block, each wave: 32 rows (2 strips) x 128 cols
//     => 16 v8f accumulators (128 VGPRs), 2 A-frags reused over 8 B-frags
//   - W chunk staged TRANSPOSED in LDS so each B fragment
//     {W[k'][col], W[k'+1][col]} is one conflict-free ds_load_b64
//   - fp32 WMMA fragment layouts per CDNA5 ISA §7.12.2:
//       A 16x4 : lanes 0-15 hold K=k,k+1 ; lanes 16-31 hold K=k+2,k+3
//       B 4x16 : v0 = B[k + 2*half][col], v1 = B[k + 2*half + 1][col]
//       C 16x16: VGPR j, lanes 0-15 -> row j ; lanes 16-31 -> row j+8
//   - requires M % 16 == 0 (true: 50000, 512)
// ---------------------------------------------------------------------------
__global__ void __launch_bounds__(128)
gemm_bn_relu_wmma(const float* __restrict__ A, const float* __restrict__ W,
                  const float* __restrict__ bias,
                  const float* __restrict__ gamma, const float* __restrict__ beta,
                  const float* __restrict__ mean,  const float* __restrict__ var,
                  float* __restrict__ out, int M, int K, int has_bn) {
    __shared__ float sWt[(size_t)D * WT_STRIDE];   // 128 cols x 132 -> 66 KB

    const int tid  = threadIdx.x;
    const int wave = tid >> 5;          // 0..3
    const int lane = tid & 31;
    const int half = lane >> 4;         // 0 | 1  (upper half-wave)
    const int l16  = lane & 15;
    const int rowBase = blockIdx.x * 128 + wave * 32;
    const bool active0 = rowBase < M;            // strip 0: rows rowBase..+15
    const bool active1 = (rowBase + 16) < M;     // strip 1: rows rowBase+16..+31

    v8f acc0[8], acc1[8];
#pragma unroll
    for (int t = 0; t < 8; ++t) {
        acc0[t] = (v8f){0.f,0.f,0.f,0.f,0.f,0.f,0.f,0.f};
        acc1[t] = (v8f){0.f,0.f,0.f,0.f,0.f,0.f,0.f,0.f};
    }

    // clamp invalid strip-1 rows onto strip 0 (loads harmless, stores guarded)
    const int row0 = rowBase + l16;
    const int row1 = active1 ? (rowBase + 16 + l16) : row0;

    for (int kk = 0; kk < K; kk += 128) {
        // stage W[kk..kk+128) x [0..128) transposed: sWt[c*WT_STRIDE + r] = W[kk+r][c]
        for (int idx = tid; idx < D * D; idx += 128) {
            int r = idx >> 7, c = idx & 127;
            sWt[(size_t)c * WT_STRIDE + r] = W[(size_t)(kk + r) * D + c];
        }
        __syncthreads();

        if (active0) {
            const float* Ar0 = A + (size_t)row0 * K + kk + half * 2;
            const float* Ar1 = A + (size_t)row1 * K + kk + half * 2;
            for (int k = 0; k < 128; k += 4) {
                v2f a0 = *(const v2f*)(Ar0 + k);
                v2f a1 = *(const v2f*)(Ar1 + k);
#pragma unroll
                for (int t = 0; t < 8; ++t) {
                    const int col = t * 16 + l16;
                    // one b64: {W[kk+k'][col], W[kk+k'+1][col]}, k' = k + 2*half
                    v2f b = *(const v2f*)(&sWt[(size_t)col * WT_STRIDE + k + half * 2]);
                    acc0[t] = __builtin_amdgcn_wmma_f32_16x16x4_f32(
                        false, a0, false, b, (short)0, acc0[t], false, false);
                    acc1[t] = __builtin_amdgcn_wmma_f32_16x16x4_f32(
                        false, a1, false, b, (short)0, acc1[t], false, false);
                }
            }
        }
        __syncthreads();
    }

    if (!active0) return;

    // epilogue: +bias, inference BN (folded scale/shift), ReLU
#pragma unroll
    for (int t = 0; t < 8; ++t) {
        const int col = t * 16 + l16;
        float bs = bias[col];
        float sc = 1.f, sh = 0.f;
        if (has_bn) {
            sc = gamma[col] * rsqrtf(var[col] + BN_EPS);
            sh = beta[col] - mean[col] * sc;
        }
#pragma unroll
        for (int j = 0; j < 8; ++j) {
            const int r0 = rowBase + half * 8 + j;
            float v = acc0[t][j] + bs;
            v = v * sc + sh;
            out[(size_t)r0 * D + col] = fmaxf(v, 0.f);
        }
        if (active1) {
#pragma unroll
            for (int j = 0; j < 8; ++j) {
                const int r1 = rowBase + 16 + half * 8 + j;
                float v = acc1[t][j] + bs;
                v = v * sc + sh;
                out[(size_t)r1 * D + col] = fmaxf(v, 0.f);
            }
        }
    }
}

// ---------------------------------------------------------------------------
// Tail: out[512,10] = hid[512,128] @ Wc2[128,10] + bc2   (tiny, plain VALU)
// ---------------------------------------------------------------------------
__global__ void classifier2_kernel(const float* __restrict__ hid,
                                   const float* __restrict__ Wc2,
                                   const float* __restrict__ bc2,
                                   float* __restrict__ out) {
    int idx = blockIdx.x * blockDim.x + threadIdx.x;
    if (idx >= N_GRAPHS * NUM_CLS) return;
    int g = idx / NUM_CLS, c = idx % NUM_CLS;
    float s = bc2[c];
    const float* hrow = hid + (size_t)g * D;
#pragma unroll 4
    for (int k = 0; k < D; ++k) s += hrow[k] * Wc2[k * NUM_CLS + c];
    out[idx] = s;
}

// ---------------------------------------------------------------------------
// Host-side launch sequence (graph-capture safe: kernels only, one stream)
// ---------------------------------------------------------------------------
extern "C" void kernel_launch(void* const* d_in, const int* in_sizes, int n_in,
                              void* d_out, int out_size, void* d_ws, size_t ws_size,
                              hipStream_t stream) {
    const float* x    = (const float*)d_in[0];
    const int*   eidx = (const int*)d_in[1];          // [2, N_EDGES]
    const int*   batch= (const int*)d_in[2];
    const float* W1s  = (const float*)d_in[3];
    const float* b1s  = (const float*)d_in[4];
    const float* g1s  = (const float*)d_in[5];
    const float* be1s = (const float*)d_in[6];
    const float* mu1s = (const float*)d_in[7];
    const float* va1s = (const float*)d_in[8];
    const float* W2s  = (const float*)d_in[9];
    const float* b2s  = (const float*)d_in[10];
    const float* gos  = (const float*)d_in[11];
    const float* beos = (const float*)d_in[12];
    const float* muos = (const float*)d_in[13];
    const float* vaos = (const float*)d_in[14];
    const float* Wc1  = (const float*)d_in[15];
    const float* bc1  = (const float*)d_in[16];
    const float* Wc2  = (const float*)d_in[17];
    const float* bc2  = (const float*)d_in[18];

    const int* e_src = eidx;
    const int* e_dst = eidx + N_EDGES;

    // workspace layout (floats)
    float* ws   = (float*)d_ws;
    float* hcur = ws;                               // [N_NODES, D]
    float* t    = hcur + (size_t)N_NODES * D;       // [N_NODES, D]
    float* t1   = t    + (size_t)N_NODES * D;       // [N_NODES, D]
    float* feat = t1   + (size_t)N_NODES * D;       // [N_GRAPHS, L*D]
    float* hid  = feat + (size_t)N_GRAPHS * L_LAYERS * D;  // [N_GRAPHS, D]

    const int featN = N_GRAPHS * L_LAYERS * D;
    zero_kernel<<<(featN + 255) / 256, 256, 0, stream>>>(feat, featN);

    const int n4        = N_NODES * D / 4;
    const int copyGrid  = (n4 + 255) / 256;
    const long long eT  = (long long)N_EDGES * 32;
    const int edgeGrid  = (int)((eT + 255) / 256);
    const long long pT  = (long long)N_NODES * 32;
    const int poolGrid  = (int)((pT + 255) / 256);
    const int gemmGridM = (N_NODES + 127) / 128;    // 128 rows / block

    const float* h = x;
    for (int i = 0; i < L_LAYERS; ++i) {
        // t = h ; t[dst] += h[src]  (GIN eps=0 aggregation)
        copy4_kernel<<<copyGrid, 256, 0, stream>>>(h, t, n4);
        edge_scatter_kernel<<<edgeGrid, 256, 0, stream>>>(e_src, e_dst, h, t);

        // t1 = relu(bn1(t @ W1 + b1))
        gemm_bn_relu_wmma<<<gemmGridM, 128, 0, stream>>>(
            t, W1s + (size_t)i * D * D, b1s + i * D,
            g1s + i * D, be1s + i * D, mu1s + i * D, va1s + i * D,
            t1, N_NODES, D, 1);

        // hcur = relu(bn_out(t1 @ W2 + b2))
        gemm_bn_relu_wmma<<<gemmGridM, 128, 0, stream>>>(
            t1, W2s + (size_t)i * D * D, b2s + i * D,
            gos + i * D, beos + i * D, muos + i * D, vaos + i * D,
            hcur, N_NODES, D, 1);

        // feat[:, i*D:(i+1)*D] += per-graph sum of hcur
        pool_kernel<<<poolGrid, 256, 0, stream>>>(hcur, batch, feat, i);
        h = hcur;
    }

    // hid = relu(feat @ Wc1 + bc1)   (K = L*D = 512, no BN)
    gemm_bn_relu_wmma<<<(N_GRAPHS + 127) / 128, 128, 0, stream>>>(
        feat, Wc1, bc1, nullptr, nullptr, nullptr, nullptr,
        hid, N_GRAPHS, L_LAYERS * D, 0);

    // out = hid @ Wc2 + bc2
    classifier2_kernel<<<(N_GRAPHS * NUM_CLS + 63) / 64, 64, 0, stream>>>(
        hid, Wc2, bc2, (float*)d_out);
}